// ModePool2d_63007170232956
// MI455X (gfx1250) — compile-verified
//
#include <hip/hip_runtime.h>
#include <hip/hip_bf16.h>

// ModePool2d (K=3, S=2, P=1) over (8,64,224,224) f32 -> (8,64,112,112) f32.
// Bandwidth-bound (~128.5 MB @ 23.3 TB/s => ~5.5us floor). Strategy:
//   - one 32x8 output tile per 256-thread (8-wave, wave32) block
//   - stage the 65x17 input halo tile into LDS with gfx1250 async
//     global->LDS copies (ASYNCcnt), zeros written for padded/OOB cells
//   - per-thread 9-sample mode via all-pairs equality counts (registers only)

#define TILE_W 32
#define TILE_H 8
#define IN_TW  (2 * TILE_W + 1)   // 65 input cols incl. halo
#define IN_TH  (2 * TILE_H + 1)   // 17 input rows incl. halo
#define LDS_STRIDE IN_TW          // 65: stride mod 64 banks == 1 -> conflict-free rows
#define LDS_ELEMS (IN_TH * LDS_STRIDE)  // 1105 floats = 4420 B per block

typedef __attribute__((address_space(1))) int global_i32;
typedef __attribute__((address_space(3))) int lds_i32;

__device__ __forceinline__ void async_copy_g2l(const float* g, float* l) {
#if defined(__gfx1250__) && __has_builtin(__builtin_amdgcn_global_load_async_to_lds_b32)
    // Builtin expects non-const AS(1)/AS(3) int*: strip const via inttoptr
    // (flat address == global address on amdgcn), addrspacecast for LDS.
    __builtin_amdgcn_global_load_async_to_lds_b32(
        (global_i32*)(unsigned long long)g,
        (lds_i32*)l,
        0, 0);
#else
    // Fallback: per-lane async copy, LDS byte offset in a VGPR, 64-bit global
    // address in a VGPR pair (GV mode, saddr=off).
    unsigned lds_off = (unsigned)(unsigned long long)(lds_i32*)l;
    asm volatile("global_load_async_to_lds_b32 %0, %1, off"
                 :: "v"(lds_off), "v"(g) : "memory");
#endif
}

__device__ __forceinline__ void wait_async_zero() {
#if defined(__gfx1250__) && __has_builtin(__builtin_amdgcn_s_wait_asynccnt)
    __builtin_amdgcn_s_wait_asynccnt(0);
#else
    asm volatile("s_wait_asynccnt 0" ::: "memory");
#endif
}

__global__ __launch_bounds__(256) void modepool2d_kernel(
    const float* __restrict__ in, float* __restrict__ out,
    int H, int W, int Ho, int Wo)
{
    __shared__ float tile[LDS_ELEMS];

    const int bc  = blockIdx.z;                 // fused batch*channel plane
    const int oh0 = blockIdx.y * TILE_H;
    const int ow0 = blockIdx.x * TILE_W;
    const int r0  = oh0 * 2 - 1;                // top input row of tile (pad=1)
    const int c0  = ow0 * 2 - 1;                // left input col of tile
    const int tid = threadIdx.y * TILE_W + threadIdx.x;

    const float* src = in + (size_t)bc * H * W;

    // ---- Stage halo tile into LDS: async copies for in-range, 0 for pad/OOB.
    #pragma unroll
    for (int i = tid; i < LDS_ELEMS; i += 256) {
        int r  = i / LDS_STRIDE;
        int c  = i - r * LDS_STRIDE;
        int gr = r0 + r;
        int gc = c0 + c;
        if (((unsigned)gr < (unsigned)H) & ((unsigned)gc < (unsigned)W)) {
            async_copy_g2l(src + (size_t)gr * W + gc, &tile[i]);
        } else {
            tile[i] = 0.0f;                     // padding contributes bin 0
        }
    }
    wait_async_zero();      // my wave's async copies landed in LDS
    __syncthreads();        // everyone's copies + zero-stores visible

    // ---- Compute one output pixel per thread from LDS.
    const int ow = ow0 + threadIdx.x;
    const int oh = oh0 + threadIdx.y;
    if (ow < Wo && oh < Ho) {
        const int lx = threadIdx.x * 2;
        const int ly = threadIdx.y * 2;

        int bins[9];
        #pragma unroll
        for (int kh = 0; kh < 3; ++kh) {
            #pragma unroll
            for (int kw = 0; kw < 3; ++kw) {
                float v = tile[(ly + kh) * LDS_STRIDE + (lx + kw)];
                float r = rintf(v * 16.0f);     // v_rndne_f32 == jnp.round
                r = fminf(fmaxf(r, 0.0f), 16.0f);
                bins[kh * 3 + kw] = (int)r;
            }
        }

        // Mode with JAX argmax tie-break (smallest bin among max counts).
        int bestCnt = 0;
        int bestBin = 17;
        #pragma unroll
        for (int i = 0; i < 9; ++i) {
            int cnt = 0;
            #pragma unroll
            for (int j = 0; j < 9; ++j) cnt += (bins[j] == bins[i]);
            bool better = (cnt > bestCnt) | ((cnt == bestCnt) & (bins[i] < bestBin));
            if (better) { bestCnt = cnt; bestBin = bins[i]; }
        }

        out[((size_t)bc * Ho + oh) * Wo + ow] = (float)bestBin * 0.0625f;
    }
}

extern "C" void kernel_launch(void* const* d_in, const int* in_sizes, int n_in,
                              void* d_out, int out_size, void* d_ws, size_t ws_size,
                              hipStream_t stream) {
    (void)in_sizes; (void)n_in; (void)out_size; (void)d_ws; (void)ws_size;

    const float* x = (const float*)d_in[0];
    float* out = (float*)d_out;

    const int B = 8, C = 64, H = 224, W = 224;
    const int Ho = 112, Wo = 112;

    dim3 block(TILE_W, TILE_H, 1);                       // 256 threads = 8 waves
    dim3 grid((Wo + TILE_W - 1) / TILE_W,                // 4
              (Ho + TILE_H - 1) / TILE_H,                // 14
              B * C);                                    // 512 planes

    modepool2d_kernel<<<grid, block, 0, stream>>>(x, out, H, W, Ho, Wo);
}